// ExponentialLookbackMoE_13134009991275
// MI455X (gfx1250) — compile-verified
//
#include <hip/hip_runtime.h>
#include <hip/hip_bf16.h>
#include <math.h>

typedef __attribute__((ext_vector_type(2))) float v2f;
typedef __attribute__((ext_vector_type(8))) float v8f;

#define SL    4096
#define DM    128
#define NK    16
#define PADX  32
#define XTROW (PADX + SL)      // 4128

// workspace layout (float offsets)
#define OFF_W     ((size_t)0)                          // 64: w[0..3], s[16..31]
#define OFF_XTP   ((size_t)64)                         // 2*128*4128
#define OFF_PEFF  (OFF_XTP + (size_t)2*DM*XTROW)       // 4*2*16*128*128
#define OFF_MUEFF (OFF_PEFF + (size_t)4*2*NK*DM*DM)    // 4*128*384
#define OFF_UC    (OFF_MUEFF + (size_t)4*DM*384)       // 2*16*128*4096

// ---------------------------------------------------------------- prep: weights & sigma^0.25
__global__ void k_prep_weights(const float* __restrict__ lw, const float* __restrict__ gum,
                               const float* __restrict__ sigma, float* __restrict__ wsw) {
    int t = threadIdx.x;
    if (t == 0) {
        float z[4]; float m = -1e30f;
        for (int i = 0; i < 4; ++i) { z[i] = lw[i] + gum[i]; m = fmaxf(m, z[i]); }
        float ex[4]; float sum = 0.f;
        for (int i = 0; i < 4; ++i) { ex[i] = expf(z[i] - m); sum += ex[i]; }
        float ys[4];
        for (int i = 0; i < 4; ++i) ys[i] = ex[i] / sum;
        int am = 0; float best = ys[0];
        for (int i = 1; i < 4; ++i) if (ys[i] > best) { best = ys[i]; am = i; }
        for (int i = 0; i < 4; ++i) {
            float h = (i == am) ? 1.0f : 0.0f;
            wsw[i] = (h - ys[i]) + ys[i];          // exact forward expression of reference
        }
    }
    if (t < NK) wsw[16 + t] = powf(sigma[t], 0.25f);
}

// ---------------------------------------------------------------- prep: x transposed + front zero-pad
__global__ void k_prep_xtp(const float* __restrict__ x, float* __restrict__ xtp) {
    int tid = blockIdx.x * 256 + threadIdx.x;
    int total = 2 * DM * XTROW;
    if (tid >= total) return;
    int tp = tid % XTROW;
    int d  = (tid / XTROW) % DM;
    int b  = tid / (XTROW * DM);
    float v = 0.0f;
    if (tp >= PADX) v = x[((size_t)b * SL + (tp - PADX)) * DM + d];
    xtp[tid] = v;
}

// ---------------------------------------------------------------- prep: effective spectral proj  P_eff[r][s][e][k*128+d]
__global__ void k_prep_peff(const float* __restrict__ Mpp, const float* __restrict__ Mpm,
                            const float* __restrict__ wsw, float* __restrict__ peff) {
    int tid = blockIdx.x * 256 + threadIdx.x;     // 4*2*16*128*128 = 2,097,152
    int d = tid & 127;
    int e = (tid >> 7) & 127;
    int k = (tid >> 14) & 15;
    int s = (tid >> 18) & 1;
    int r = tid >> 19;
    const float* M = s ? Mpm : Mpp;
    float acc = 0.0f;
    for (int i = 0; i <= r; ++i)
        acc += wsw[i] * M[((size_t)(i * NK + k) * DM + d) * DM + e];
    acc *= wsw[16 + k];                            // sigma^0.25 folded in
    peff[(((size_t)(r * 2 + s) * DM + e) << 11) + (k * DM + d)] = acc;
}

// ---------------------------------------------------------------- prep: effective AR proj  Mu_eff[r][e][j*128+d]
__global__ void k_prep_mueff(const float* __restrict__ Mu, const float* __restrict__ wsw,
                             float* __restrict__ mueff) {
    int tid = blockIdx.x * 256 + threadIdx.x;     // 4*128*384 = 196,608
    int jd = tid % 384;
    int e  = (tid / 384) % DM;
    int r  = tid / (384 * DM);
    int j = jd >> 7, d = jd & 127;
    float acc = 0.0f;
    for (int i = 0; i <= r; ++i)
        acc += wsw[i] * Mu[((size_t)(i * 3 + j) * DM + d) * DM + e];
    mueff[((size_t)(r * DM + e)) * 384 + jd] = acc;
}

// ---------------------------------------------------------------- big conv: Uc[s][k][d][t] = sum_j phi_s[j,k] x[t-j,d]
// wave owns TWO adjacent 16-row t-blocks (B tiles shared), both signs -> 4 WMMAs per lag chunk.
__global__ __launch_bounds__(256) void k_conv(const float* __restrict__ xtp,
                                              const float* __restrict__ phi,
                                              float* __restrict__ uc, int b) {
    __shared__ float sx[XTROW];                    // 16.5 KB staged x column prefix (incl. 32 zero pad)
    int d   = blockIdx.x;                          // 0..127
    int seg = blockIdx.y;                          // 0..15 (each = 8 waves * 32 t-rows = 256 t)
    int nst = PADX + (seg + 1) * 256;
    const float* xrow = xtp + ((size_t)b * DM + d) * XTROW;

    // CDNA5 async copy global->LDS (ASYNCcnt), then drain + workgroup barrier
    for (int i = threadIdx.x; i < nst; i += 256) {
        uint32_t lp = (uint32_t)(uintptr_t)(sx + i);
        const float* gp = xrow + i;
        asm volatile("global_load_async_to_lds_b32 %0, %1, off"
                     :: "v"(lp), "v"(gp) : "memory");
    }
    asm volatile("s_wait_asynccnt 0x0" ::: "memory");
    __syncthreads();

    __builtin_prefetch(phi, 0, 1);                 // global_prefetch_b8: warm phi stream

    int lane = threadIdx.x & 31;
    int wv   = __builtin_amdgcn_readfirstlane(threadIdx.x >> 5);   // scalar wave id -> uniform loop bound
    int half = lane >> 4;
    int ln   = lane & 15;
    int T0   = seg * 256 + wv * 32;                // lower t-block; upper is T0+16
    int abase = PADX + T0 + ln - 2 * half;         // A[m,kk]=x[T0+m-J0-kk]; VGPR0: kk=2*half

    v8f aap = {}, aam = {}, abp = {}, abm = {};
    int jend = T0 + 16 + 12;                       // covers j up to T0+31 (upper block causal bound)
#pragma unroll 4
    for (int J0 = 0; J0 <= jend; J0 += 4) {
        v2f Aa, Ab;
        Aa.x = sx[abase - J0];                     // kk = 2*half      (lower block)
        Aa.y = sx[abase - J0 - 1];                 // kk = 2*half+1
        Ab.x = sx[abase - J0 + 16];                // upper block (t+16)
        Ab.y = sx[abase - J0 + 15];
        int pidx = (J0 + 2 * half) * NK + ln;      // B[kk,n] = phi[J0+kk, n]
        v2f Bp, Bm;
        Bp.x = phi[pidx];
        Bp.y = phi[pidx + NK];
        Bm.x = Bp.x;                               // (-1)^(J0+2h)   = +1  (J0%4==0)
        Bm.y = -Bp.y;                              // (-1)^(J0+2h+1) = -1
        aap = __builtin_amdgcn_wmma_f32_16x16x4_f32(false, Aa, false, Bp, (short)0, aap, false, false);
        aam = __builtin_amdgcn_wmma_f32_16x16x4_f32(false, Aa, false, Bm, (short)0, aam, false, false);
        abp = __builtin_amdgcn_wmma_f32_16x16x4_f32(false, Ab, false, Bp, (short)0, abp, false, false);
        abm = __builtin_amdgcn_wmma_f32_16x16x4_f32(false, Ab, false, Bm, (short)0, abm, false, false);
    }
    // D layout: lane ln = k, value v -> t = T0 + 8*half + v  (8 consecutive t per lane)
    size_t base  = (((size_t)ln) * DM + d) * SL + T0 + 8 * half;   // sign 0, lower block
    size_t soff  = (size_t)NK * DM * SL;                           // sign stride
    float4* p;
    p = (float4*)(uc + base);
    p[0] = make_float4(aap[0], aap[1], aap[2], aap[3]);
    p[1] = make_float4(aap[4], aap[5], aap[6], aap[7]);
    p = (float4*)(uc + base + 16);
    p[0] = make_float4(abp[0], abp[1], abp[2], abp[3]);
    p[1] = make_float4(abp[4], abp[5], abp[6], abp[7]);
    p = (float4*)(uc + base + soff);
    p[0] = make_float4(aam[0], aam[1], aam[2], aam[3]);
    p[1] = make_float4(aam[4], aam[5], aam[6], aam[7]);
    p = (float4*)(uc + base + soff + 16);
    p[0] = make_float4(abm[0], abm[1], abm[2], abm[3]);
    p[1] = make_float4(abm[4], abm[5], abm[6], abm[7]);
}

// ---------------------------------------------------------------- projection + AR -> out[b][t][e]
// wave owns a t-PAIR (A tiles shared across the two t-blocks) and an e-quarter (2 e-tiles).
__global__ __launch_bounds__(256) void k_proj(const float* __restrict__ uc,
                                              const float* __restrict__ peff,
                                              const float* __restrict__ mueff,
                                              const float* __restrict__ xtp,
                                              float* __restrict__ out, int b) {
    int lane = threadIdx.x & 31;
    int wv   = __builtin_amdgcn_readfirstlane(threadIdx.x >> 5);
    int half = lane >> 4;
    int ln   = lane & 15;
    int tp = blockIdx.x * 8 + wv;                  // 0..127 t-pairs
    int T0 = tp * 32;                              // blocks T0 and T0+16 (never straddle a 512 range)
    int r = (T0 >= 2048) ? 0 : (T0 >= 1024) ? 1 : (T0 >= 512) ? 2 : 3;
    int E0 = blockIdx.y * 32;                      // 2 e-tiles of 16

    v8f a00 = {}, a01 = {}, a10 = {}, a11 = {};    // [etile][tblock]

    // spectral: contraction over kd = k*128+d (2048), both signs
    for (int s = 0; s < 2; ++s) {
        const float* P = peff + (((size_t)(r * 2 + s) * DM) << 11);
        const float* U = uc + (size_t)s * NK * DM * SL;
        __builtin_prefetch(P + (((size_t)(E0 + ln)) << 11), 0, 1);
#pragma unroll 2
        for (int C0 = 0; C0 < 2048; C0 += 4) {
            const float* up = U + (size_t)(C0 + 2 * half) * SL + T0 + ln;
            v2f B0, B1;
            B0.x = up[0];   B0.y = up[SL];                             // rows kk, kk+1 @ t-block 0
            B1.x = up[16];  B1.y = up[SL + 16];                        // @ t-block 1
            const float* pr = P + (((size_t)(E0 + ln)) << 11) + C0 + 2 * half;
            v2f A0, A1;
            A0.x = pr[0];     A0.y = pr[1];
            A1.x = pr[32768]; A1.y = pr[32769];                        // e+16 -> +16*2048
            a00 = __builtin_amdgcn_wmma_f32_16x16x4_f32(false, A0, false, B0, (short)0, a00, false, false);
            a01 = __builtin_amdgcn_wmma_f32_16x16x4_f32(false, A0, false, B1, (short)0, a01, false, false);
            a10 = __builtin_amdgcn_wmma_f32_16x16x4_f32(false, A1, false, B0, (short)0, a10, false, false);
            a11 = __builtin_amdgcn_wmma_f32_16x16x4_f32(false, A1, false, B1, (short)0, a11, false, false);
        }
    }
    // AR: contraction over jd = j*128+d (384); B rows from shifted x columns
    {
        const float* Xb = xtp + (size_t)b * DM * XTROW;
        const float* Mb = mueff + (size_t)r * DM * 384;
#pragma unroll 2
        for (int C0 = 0; C0 < 384; C0 += 4) {
            int j  = C0 >> 7;
            int d0 = C0 & 127;
            const float* bp = Xb + (size_t)(d0 + 2 * half) * XTROW + PADX + T0 + ln - j;
            v2f B0, B1;
            B0.x = bp[0];   B0.y = bp[XTROW];
            B1.x = bp[16];  B1.y = bp[XTROW + 16];
            const float* mr = Mb + (size_t)(E0 + ln) * 384 + C0 + 2 * half;
            v2f A0, A1;
            A0.x = mr[0];    A0.y = mr[1];
            A1.x = mr[6144]; A1.y = mr[6145];                          // e+16 -> +16*384
            a00 = __builtin_amdgcn_wmma_f32_16x16x4_f32(false, A0, false, B0, (short)0, a00, false, false);
            a01 = __builtin_amdgcn_wmma_f32_16x16x4_f32(false, A0, false, B1, (short)0, a01, false, false);
            a10 = __builtin_amdgcn_wmma_f32_16x16x4_f32(false, A1, false, B0, (short)0, a10, false, false);
            a11 = __builtin_amdgcn_wmma_f32_16x16x4_f32(false, A1, false, B1, (short)0, a11, false, false);
        }
    }
    // D layout: m=e (rel = v+8*half), n=t (rel = ln); per lane 8 consecutive e
    float* ob0 = out + ((size_t)b * SL + T0 + ln) * DM + E0 + 8 * half;
    float* ob1 = ob0 + (size_t)16 * DM;            // second t-block
    *(float4*)(ob0 +  0) = make_float4(a00[0], a00[1], a00[2], a00[3]);
    *(float4*)(ob0 +  4) = make_float4(a00[4], a00[5], a00[6], a00[7]);
    *(float4*)(ob0 + 16) = make_float4(a10[0], a10[1], a10[2], a10[3]);
    *(float4*)(ob0 + 20) = make_float4(a10[4], a10[5], a10[6], a10[7]);
    *(float4*)(ob1 +  0) = make_float4(a01[0], a01[1], a01[2], a01[3]);
    *(float4*)(ob1 +  4) = make_float4(a01[4], a01[5], a01[6], a01[7]);
    *(float4*)(ob1 + 16) = make_float4(a11[0], a11[1], a11[2], a11[3]);
    *(float4*)(ob1 + 20) = make_float4(a11[4], a11[5], a11[6], a11[7]);
}

// ----------------------------------------------------------------
extern "C" void kernel_launch(void* const* d_in, const int* in_sizes, int n_in,
                              void* d_out, int out_size, void* d_ws, size_t ws_size,
                              hipStream_t stream) {
    (void)in_sizes; (void)n_in; (void)out_size; (void)ws_size;
    const float* x     = (const float*)d_in[0];
    const float* sigma = (const float*)d_in[1];
    const float* phi   = (const float*)d_in[2];
    const float* Mu    = (const float*)d_in[3];
    const float* Mpp   = (const float*)d_in[4];
    const float* Mpm   = (const float*)d_in[5];
    const float* lw    = (const float*)d_in[6];
    const float* gum   = (const float*)d_in[7];
    float* out = (float*)d_out;
    float* ws  = (float*)d_ws;

    float* wsw   = ws + OFF_W;
    float* xtp   = ws + OFF_XTP;
    float* peff  = ws + OFF_PEFF;
    float* mueff = ws + OFF_MUEFF;
    float* uc    = ws + OFF_UC;

    k_prep_weights<<<1, 32, 0, stream>>>(lw, gum, sigma, wsw);
    k_prep_xtp<<<(2 * DM * XTROW + 255) / 256, 256, 0, stream>>>(x, xtp);
    k_prep_peff<<<(4 * 2 * NK * DM * DM) / 256, 256, 0, stream>>>(Mpp, Mpm, wsw, peff);
    k_prep_mueff<<<(4 * DM * 384) / 256, 256, 0, stream>>>(Mu, wsw, mueff);

    for (int b = 0; b < 2; ++b) {
        k_conv<<<dim3(DM, 16), 256, 0, stream>>>(xtp, phi, uc, b);
        k_proj<<<dim3(16, 4), 256, 0, stream>>>(uc, peff, mueff, xtp, out, b);
    }
}